// PatchLevelPruner_33638183863200
// MI455X (gfx1250) — compile-verified
//
#include <hip/hip_runtime.h>
#include <hip/hip_bf16.h>
#include <cstdint>

typedef float v2f __attribute__((ext_vector_type(2)));
typedef float v8f __attribute__((ext_vector_type(8)));

#define PS      4
#define HH      128
#define WW      128
#define CC      256
#define CH      64
#define PNUM    1024      // patches per image (32x32)
#define BATCH   16
#define KKEEP   716       // top-k
#define KH      23
#define KW      32
#define QTOT    736       // KH*KW slots (20 zero-pad)
#define OUT_TOK 11776     // KH*PS * KW*PS = 92*128

__device__ __forceinline__ float gelu_tanh(float x) {
    // jax.nn.gelu default (tanh approximation)
    float x3 = x * x * x;
    float t  = tanhf(0.7978845608028654f * (x + 0.044715f * x3));
    return 0.5f * x * (1.0f + t);
}
__device__ __forceinline__ float sigmoidf_(float x) {
    return 1.0f / (1.0f + __expf(-x));
}

// ---------------------------------------------------------------------------
// Kernel A: per-patch importance. One wave32 per patch: 16 tokens == one WMMA
// M-tile. D = A(16x4 f32) x B(4x16 f32) + C accumulated over K=256, N=64.
//
// LDS holds w1 pre-interleaved by K-pairs with an XOR-32 bank swizzle:
//   w1t[kp*128 + ((2*col) ^ ((kp&1)<<5)) + (k&1)] = w1[k*64 + col],  kp = k>>1
// so each lane's B operand {B[k][n], B[k+1][n]} is one aligned ds_load_b64,
// and the two lane-halves (even/odd kp) hit disjoint bank sets.
// ---------------------------------------------------------------------------
__global__ __launch_bounds__(256) void imp_kernel(
    const float* __restrict__ tokens,
    const float* __restrict__ w1,
    const float* __restrict__ b1,
    const float* __restrict__ w2,
    const float* __restrict__ b2,
    float* __restrict__ patch_imp)
{
    __shared__ float w1t[CC * CH];   // 64 KB of the 320 KB/WGP LDS

    // Stage + transform w1 (coalesced global reads, scattered LDS writes, once)
    for (int i = threadIdx.x; i < CC * CH; i += 256) {
        const int k   = i >> 6;          // 0..255
        const int col = i & 63;          // 0..63
        const int kp  = k >> 1;
        const int pos = kp * 128 + (((col << 1) ^ ((kp & 1) << 5)) | (k & 1));
        w1t[pos] = w1[i];
    }
    __syncthreads();

    const int lane    = threadIdx.x & 31;
    const int wave    = threadIdx.x >> 5;
    const int patchId = blockIdx.x * 8 + wave;        // [0, BATCH*PNUM)
    const int b       = patchId >> 10;
    const int p       = patchId & (PNUM - 1);
    const int ph      = p >> 5, pw = p & 31;

    const int m    = lane & 15;     // M row (token within patch) / N col
    const int half = lane >> 4;     // selects K sub-pair per ISA f32 layout

    // token (i,j) inside the 4x4 patch for this lane's M row
    const int row = (ph * PS + (m >> 2)) * WW + pw * PS + (m & 3);
    const float* tokrow = tokens + ((size_t)b * (HH * WW) + row) * CC;

    v8f acc[4];
    #pragma unroll
    for (int nt = 0; nt < 4; ++nt) acc[nt] = (v8f){0.f,0.f,0.f,0.f,0.f,0.f,0.f,0.f};

    #pragma unroll 4
    for (int ks = 0; ks < CC / 4; ++ks) {
        const int k0 = ks * 4 + half * 2;            // K base for this lane half
        const int kp = 2 * ks + half;                // K-pair index
        v2f av = *(const v2f*)(tokrow + k0);         // {A[m][k0], A[m][k0+1]}
        const float* wrow = &w1t[kp * 128];
        const int sw = (kp & 1) << 5;
        #pragma unroll
        for (int nt = 0; nt < 4; ++nt) {
            const int col = nt * 16 + m;
            v2f bv = *(const v2f*)(wrow + (((col << 1) ^ sw)));  // {B[k0][n], B[k0+1][n]}
            acc[nt] = __builtin_amdgcn_wmma_f32_16x16x4_f32(
                false, av, false, bv, (short)0, acc[nt], false, false);
        }
    }

    // Epilogue: + b1, gelu, dot with w2 (per-lane channel), + b2, sigmoid, mean
    float b1r[4], w2r[4];
    #pragma unroll
    for (int nt = 0; nt < 4; ++nt) {
        b1r[nt] = b1[nt * 16 + m];
        w2r[nt] = w2[nt * 16 + m];
    }
    const float b2v = b2[0];

    // acc[nt][r] = token (r + 8*half), channel (nt*16 + m)
    float partial[8];
    #pragma unroll
    for (int r = 0; r < 8; ++r) {
        float s = 0.f;
        #pragma unroll
        for (int nt = 0; nt < 4; ++nt) {
            float h = acc[nt][r] + b1r[nt];
            s += gelu_tanh(h) * w2r[nt];
        }
        partial[r] = s;
    }
    // reduce channels across the 16 lanes of each half
    #pragma unroll
    for (int r = 0; r < 8; ++r) {
        #pragma unroll
        for (int mask = 1; mask < 16; mask <<= 1)
            partial[r] += __shfl_xor(partial[r], mask, 32);
    }
    float acc_imp = 0.f;
    #pragma unroll
    for (int r = 0; r < 8; ++r) acc_imp += sigmoidf_(partial[r] + b2v);
    // combine the two token halves (tokens 0-7 vs 8-15)
    acc_imp += __shfl_xor(acc_imp, 16, 32);
    if (lane == 0) patch_imp[patchId] = acc_imp * (1.0f / 16.0f);
}

// ---------------------------------------------------------------------------
// Kernel B: exact top-k (k=716) with ascending-index output. One block/batch.
// rank(i) = #{j : v[j] > v[i]  or (v[j]==v[i] && j<i)}; keep if rank < k.
// Compaction position via inclusive LDS scan -> indices already sorted.
// ---------------------------------------------------------------------------
__global__ __launch_bounds__(1024) void select_kernel(
    const float* __restrict__ patch_imp, int* __restrict__ top_idx)
{
    __shared__ float vals[PNUM];
    __shared__ int   psum[PNUM];
    const int b   = blockIdx.x;
    const int tid = threadIdx.x;

    vals[tid] = patch_imp[b * PNUM + tid];
    __syncthreads();

    const float v = vals[tid];
    int rank = 0;
    #pragma unroll 8
    for (int j = 0; j < PNUM; ++j) {
        const float u = vals[j];
        rank += (u > v) || ((u == v) && (j < tid));
    }
    const int keep = (rank < KKEEP) ? 1 : 0;
    psum[tid] = keep;
    __syncthreads();

    // Hillis-Steele inclusive scan over 1024 flags
    for (int off = 1; off < PNUM; off <<= 1) {
        int t = (tid >= off) ? psum[tid - off] : 0;
        __syncthreads();
        psum[tid] += t;
        __syncthreads();
    }

    if (keep) top_idx[b * QTOT + psum[tid] - 1] = tid;
    if (tid < QTOT - KKEEP) top_idx[b * QTOT + KKEEP + tid] = -1;  // pad slots
}

// ---------------------------------------------------------------------------
// Kernel C: gather kept patches and repack to [B, 92, 128, 256]. float4 copy.
// total float4 = BATCH * OUT_TOK * 64 = 12,058,624 (divides 256 exactly)
// ---------------------------------------------------------------------------
__global__ __launch_bounds__(256) void gather_kernel(
    const float* __restrict__ tokens,
    const int* __restrict__ top_idx,
    float* __restrict__ out)
{
    const long long idx = (long long)blockIdx.x * 256 + threadIdx.x; // float4 idx
    const int c4  = (int)(idx & 63);
    const int tok = (int)(idx >> 6);          // [0, BATCH*OUT_TOK)
    const int b   = tok / OUT_TOK;
    const int t   = tok - b * OUT_TOK;
    const int rr  = t >> 7;                   // output row (0..91), 128 cols
    const int cc  = t & 127;
    const int q   = (rr >> 2) * KW + (cc >> 2);   // kept-patch slot
    const int i   = rr & 3, j = cc & 3;

    const int sp = top_idx[b * QTOT + q];
    float4 val = make_float4(0.f, 0.f, 0.f, 0.f);
    if (sp >= 0) {
        const int sph = sp >> 5, spw = sp & 31;
        const int srow = (sph * PS + i) * WW + spw * PS + j;
        val = *(const float4*)(tokens + ((size_t)b * (HH * WW) + srow) * CC + c4 * 4);
    }
    ((float4*)out)[idx] = val;
}

// ---------------------------------------------------------------------------
extern "C" void kernel_launch(void* const* d_in, const int* in_sizes, int n_in,
                              void* d_out, int out_size, void* d_ws, size_t ws_size,
                              hipStream_t stream)
{
    const float* tokens = (const float*)d_in[0];
    const float* w1     = (const float*)d_in[1];
    const float* b1     = (const float*)d_in[2];
    const float* w2     = (const float*)d_in[3];
    const float* b2     = (const float*)d_in[4];
    // d_in[5] = H (128), d_in[6] = W (128) — compile-time constants here.

    float* patch_imp = (float*)d_ws;                                   // 64 KB
    int*   top_idx   = (int*)((char*)d_ws + BATCH * PNUM * sizeof(float)); // 46 KB

    imp_kernel<<<BATCH * PNUM / 8, 256, 0, stream>>>(tokens, w1, b1, w2, b2, patch_imp);
    select_kernel<<<BATCH, 1024, 0, stream>>>(patch_imp, top_idx);

    const long long total_vec = (long long)BATCH * OUT_TOK * (CC / 4); // 12,058,624
    gather_kernel<<<(int)(total_vec / 256), 256, 0, stream>>>(tokens, top_idx, (float*)d_out);
}